// AucJiterator_49847390437820
// MI455X (gfx1250) — compile-verified
//
#include <hip/hip_runtime.h>
#include <stdint.h>

// ---------------------------------------------------------------------------
// Weighted ROC-AUC, 32 tasks x 1M elements, exact via two-level binning.
//   area = sum over (pos i, neg j) pairs with p_i sorted before p_j of w_i*w_j
//   AUC  = area / (Wpos*Wneg), 0.5 if denom==0
// Cross-bin term uses a WMMA (v_wmma_f32_16x16x4_f32) tensor-core prefix scan.
// ---------------------------------------------------------------------------

typedef __attribute__((ext_vector_type(2))) float v2f;
typedef __attribute__((ext_vector_type(8))) float v8f;

__device__ __forceinline__ uint32_t ordkey(float f) {
  // monotonic float -> uint mapping (bigger float => bigger uint)
  uint32_t b = __float_as_uint(f);
  uint32_t m = (uint32_t)((int32_t)b >> 31) | 0x80000000u;
  return b ^ m;
}

// ---------------- zero bins + accumulators ----------------
__global__ void zero_kernel(float* wpos, float* wneg, uint32_t* cnt,
                            double* scal, int nbin) {
  int i = blockIdx.x * blockDim.x + threadIdx.x;
  if (i < nbin) { wpos[i] = 0.0f; wneg[i] = 0.0f; cnt[i] = 0u; }
  if (i == 0) { scal[0] = 0.0; scal[1] = 0.0; scal[2] = 0.0; }
}

// ---------------- per-bin weighted histogram ----------------
__global__ void hist_kernel(const float* __restrict__ pred,
                            const float* __restrict__ lab,
                            const float* __restrict__ wt, int n,
                            float* wpos, float* wneg, uint32_t* cnt, int shift) {
  int i = blockIdx.x * blockDim.x + threadIdx.x;
  if (i >= n) return;
  uint32_t u = ordkey(pred[i]);
  int b = (int)(u >> shift);
  float l = lab[i];
  float w = wt[i];
  atomicAdd(&wpos[b], w * l);
  atomicAdd(&wneg[b], w * (1.0f - l));
  atomicAdd(&cnt[b], 1u);
}

// ---------------- WMMA-based inclusive scan of 1024 floats in LDS ----------
// Layout: value t lives at buf[t]; chunk c = t/16 (a column), k = t%16 (a row).
// Per 256-tile (one wave): Z = L * Y with L = 16x16 lower-triangular ones,
// done as 4 accumulating V_WMMA_F32_16X16X4_F32 (A = 16x4 slab of L,
// B = 4x16 slab of Y). Column totals (row 15) are then scanned serially and
// added back as chunk offsets.
__device__ void scan1024_wmma(float* buf, float* tot64, float* off64) {
  const int tid  = threadIdx.x;
  const int wave = tid >> 5;
  const int lane = tid & 31;
  const int half = lane >> 4;   // 0: lanes 0-15, 1: lanes 16-31
  const int ml   = lane & 15;   // A-row index / B-column index

  if (wave < 4) {               // wave-uniform branch: EXEC stays all-ones
    float* base = buf + wave * 256;
    v8f acc = {0.f, 0.f, 0.f, 0.f, 0.f, 0.f, 0.f, 0.f};
#pragma unroll
    for (int kk = 0; kk < 4; ++kk) {
      // A 16x4 f32 layout: lanes 0-15 hold K={0,1}, lanes 16-31 hold K={2,3}
      const int k0 = kk * 4 + (half ? 2 : 0);
      const int k1 = kk * 4 + (half ? 3 : 1);
      v2f a, b;
      a.x = (k0 <= ml) ? 1.0f : 0.0f;   // L[ml][k0]
      a.y = (k1 <= ml) ? 1.0f : 0.0f;   // L[ml][k1]
      // B 4x16 f32: row k, column ml -> Y[k][c] = buf[c*16 + k]
      b.x = base[ml * 16 + k0];
      b.y = base[ml * 16 + k1];
      acc = __builtin_amdgcn_wmma_f32_16x16x4_f32(
          /*neg_a=*/false, a, /*neg_b=*/false, b,
          /*c_mod=*/(short)0, acc, /*reuse_a=*/false, /*reuse_b=*/false);
    }
    // chunk totals = row 15 of Z (held by upper-half lanes, VGPR 7)
    if (half == 1) tot64[wave * 16 + ml] = acc[7];
    // write within-chunk inclusive values back
    const int rowBase = half * 8;
#pragma unroll
    for (int r = 0; r < 8; ++r) base[ml * 16 + rowBase + r] = acc[r];
  }
  __syncthreads();
  if (tid == 0) {               // tiny serial scan of 64 chunk totals
    float run = 0.0f;
    for (int i = 0; i < 64; ++i) { off64[i] = run; run += tot64[i]; }
  }
  __syncthreads();
  if (wave < 4) {
    const float o = off64[wave * 16 + ml];
    const int rowBase = half * 8;
#pragma unroll
    for (int r = 0; r < 8; ++r)
      buf[wave * 256 + ml * 16 + rowBase + r] += o;
  }
  __syncthreads();
}

// ---------------- scan bins: cross-bin area term + scatter offsets ---------
__global__ __launch_bounds__(1024)
void scan_kernel(const float* __restrict__ binWpos,
                 const float* __restrict__ binWneg,
                 const uint32_t* __restrict__ binCnt,
                 uint32_t* binOffset, uint32_t* binCursor,
                 double* scal, int bpt /* bins per thread = nbin/1024 */) {
  __shared__ float buf[1024];
  __shared__ float tot64[64];
  __shared__ float off64[64];
  const int tid  = threadIdx.x;
  const int base = tid * bpt;

  // per-thread partial sums over contiguous bin range (counts exact in f32:
  // all partial sums are integers < 2^24)
  float sC = 0.f, sP = 0.f, sN = 0.f;
  for (int i = 0; i < bpt; ++i) {
    const int b = base + i;
    sC += (float)binCnt[b];
    sP += binWpos[b];
    sN += binWneg[b];
  }

  buf[tid] = sC; __syncthreads();
  scan1024_wmma(buf, tot64, off64);
  const float inclC = buf[tid];
  __syncthreads();

  buf[tid] = sP; __syncthreads();
  scan1024_wmma(buf, tot64, off64);
  const float inclP   = buf[tid];
  const float wposTot = buf[1023];
  __syncthreads();

  buf[tid] = sN; __syncthreads();
  scan1024_wmma(buf, tot64, off64);
  const float wnegTot = buf[1023];

  // cross-bin term: walk my bins from highest key down, carrying the sum of
  // positive weights strictly above the current bin
  float run = wposTot - inclP;  // wpos in all bins after my range
  float cross = 0.0f;
  for (int i = bpt - 1; i >= 0; --i) {
    const int b = base + i;
    cross += binWneg[b] * run;
    run   += binWpos[b];
  }
  atomicAdd(&scal[0], (double)cross);

  // ascending exclusive count prefix -> scatter offsets; zero cursors
  uint32_t off = (uint32_t)(inclC - sC + 0.5f);
  for (int i = 0; i < bpt; ++i) {
    const int b = base + i;
    binOffset[b] = off;
    binCursor[b] = 0u;
    off += binCnt[b];
  }
  if (tid == 0) { scal[1] = (double)wposTot; scal[2] = (double)wnegTot; }
}

// ---------------- scatter records into bins ----------------
__global__ void scatter_kernel(const float* __restrict__ pred,
                               const float* __restrict__ lab,
                               const float* __restrict__ wt, int n,
                               const uint32_t* __restrict__ binOffset,
                               uint32_t* binCursor,
                               uint32_t* skey, uint32_t* swl, int shift) {
  int i = blockIdx.x * blockDim.x + threadIdx.x;
  if (i >= n) return;
  const uint32_t u = ordkey(pred[i]);
  const int b = (int)(u >> shift);
  const uint32_t pos = binOffset[b] + atomicAdd(&binCursor[b], 1u);
  uint32_t wb = __float_as_uint(wt[i]) & 0x7FFFFFFFu;  // weights in [0,1)
  if (lab[i] > 0.5f) wb |= 0x80000000u;                // label in sign bit
  skey[pos] = u;
  swl[pos]  = wb;
}

// ---------------- exact within-bin pairwise term (one wave per bin) --------
__global__ void within_kernel(const uint32_t* __restrict__ binCnt,
                              const uint32_t* __restrict__ binOffset,
                              const uint32_t* __restrict__ skey,
                              const uint32_t* __restrict__ swl,
                              double* scal, int nbin) {
  const int wavesPerBlock = blockDim.x >> 5;
  const int bin  = blockIdx.x * wavesPerBlock + (int)(threadIdx.x >> 5);
  const int lane = threadIdx.x & 31;
  if (bin >= nbin) return;  // wave-uniform
  const int c = (int)binCnt[bin];
  float acc = 0.0f;
  if (c > 1) {
    const uint32_t off = binOffset[bin];
    for (int i = lane; i < c; i += 32) {
      const uint32_t wli = swl[off + i];
      if (!(wli >> 31)) continue;                 // i ranges over positives
      const float    wi = __uint_as_float(wli & 0x7FFFFFFFu);
      const uint32_t ui = skey[off + i];
      for (int j = 0; j < c; ++j) {
        const uint32_t wlj = swl[off + j];
        if (wlj >> 31) continue;                  // j ranges over negatives
        const float    wj = __uint_as_float(wlj & 0x7FFFFFFFu);
        const uint32_t uj = skey[off + j];
        if (ui > uj)       acc += wi * wj;
        else if (ui == uj) acc += 0.5f * wi * wj; // measure-zero tie handling
      }
    }
  }
  for (int o = 16; o > 0; o >>= 1) acc += __shfl_down(acc, o, 32);
  if (lane == 0 && acc != 0.0f) atomicAdd(&scal[0], (double)acc);
}

// ---------------- finalize one task ----------------
__global__ void finalize_kernel(const double* __restrict__ scal, float* out) {
  const double denom = scal[1] * scal[2];
  *out = (denom == 0.0) ? 0.5f : (float)(scal[0] / denom);
}

// ---------------------------------------------------------------------------
extern "C" void kernel_launch(void* const* d_in, const int* in_sizes, int n_in,
                              void* d_out, int out_size, void* d_ws, size_t ws_size,
                              hipStream_t stream) {
  (void)n_in;
  const float* pred = (const float*)d_in[1];
  const float* lab  = (const float*)d_in[2];
  const float* wt   = (const float*)d_in[3];
  float* out = (float*)d_out;

  const int T = out_size;                         // 32 tasks
  const int N = (int)((long long)in_sizes[1] / (T > 0 ? T : 1));  // 1,000,000

  // Choose the largest bin count that fits the workspace (2^10 .. 2^18).
  int nbLog = 18;
  while (nbLog > 10) {
    const size_t nb = (size_t)1 << nbLog;
    const size_t need = 64 + nb * 4 * 5 + (size_t)N * 4 * 2;
    if (need <= ws_size) break;
    --nbLog;
  }
  const int nbin  = 1 << nbLog;
  const int shift = 32 - nbLog;
  const int bpt   = nbin / 1024;

  char* w = (char*)d_ws;
  double*   scal      = (double*)w;              w += 64;  // area, Wpos, Wneg
  float*    binWpos   = (float*)w;               w += (size_t)nbin * 4;
  float*    binWneg   = (float*)w;               w += (size_t)nbin * 4;
  uint32_t* binCnt    = (uint32_t*)w;            w += (size_t)nbin * 4;
  uint32_t* binOffset = (uint32_t*)w;            w += (size_t)nbin * 4;
  uint32_t* binCursor = (uint32_t*)w;            w += (size_t)nbin * 4;
  uint32_t* skey      = (uint32_t*)w;            w += (size_t)N * 4;
  uint32_t* swl       = (uint32_t*)w;

  const int gN = (N + 255) / 256;
  const int gB = (nbin + 255) / 256;
  const int gW = (nbin + 7) / 8;   // within: 8 waves per 256-thread block

  for (int t = 0; t < T; ++t) {
    const float* p = pred + (size_t)t * N;
    const float* l = lab  + (size_t)t * N;
    const float* g = wt   + (size_t)t * N;
    zero_kernel<<<gB, 256, 0, stream>>>(binWpos, binWneg, binCnt, scal, nbin);
    hist_kernel<<<gN, 256, 0, stream>>>(p, l, g, N, binWpos, binWneg, binCnt, shift);
    scan_kernel<<<1, 1024, 0, stream>>>(binWpos, binWneg, binCnt,
                                        binOffset, binCursor, scal, bpt);
    scatter_kernel<<<gN, 256, 0, stream>>>(p, l, g, N, binOffset, binCursor,
                                           skey, swl, shift);
    within_kernel<<<gW, 256, 0, stream>>>(binCnt, binOffset, skey, swl, scal, nbin);
    finalize_kernel<<<1, 1, 0, stream>>>(scal, out + t);
  }
}